// LiftSplatShoot_62010737819944
// MI455X (gfx1250) — compile-verified
//
#include <hip/hip_runtime.h>
#include <hip/hip_bf16.h>
#include <math.h>

// ---- problem constants (from reference) ----
#define BATCH       4
#define DPL         41            // depth bins
#define NCH         64            // camera channels
#define FH          32
#define FW          88
#define NPIX_PER_B  (FH*FW)       // 2816
#define TOTCH       (DPL+NCH)     // 105
#define TP          16            // pixels per block (one WMMA M-tile)
#define TILES_PER_B (NPIX_PER_B/TP) // 176
#define NX0         200
#define NZ          200
#define CH_STRIDE   (NX0*NZ)      // 40000
#define OUT_PER_B   (NCH*CH_STRIDE) // 2,560,000
#define NPTS_TILE   (DPL*TP)      // 656
#define FPITCH      66            // LDS feature pitch (bank-conflict free both ways)

typedef __attribute__((ext_vector_type(2))) float v2f;
typedef __attribute__((ext_vector_type(8))) float v8f;

__global__ void zero_kernel(float4* __restrict__ out, int n4) {
    int i = blockIdx.x * blockDim.x + threadIdx.x;
    if (i < n4) {
        float4 z; z.x = 0.f; z.y = 0.f; z.z = 0.f; z.w = 0.f;
        out[i] = z;
    }
}

__device__ inline void inv3(const float* __restrict__ m, float* o) {
    float a = m[0], b = m[1], c = m[2];
    float d = m[3], e = m[4], f = m[5];
    float g = m[6], h = m[7], i = m[8];
    float det = a*(e*i - f*h) - b*(d*i - f*g) + c*(d*h - e*g);
    o[0] =  (e*i - f*h) / det;
    o[1] = -(b*i - c*h) / det;
    o[2] =  (b*f - c*e) / det;
    o[3] = -(d*i - f*g) / det;
    o[4] =  (a*i - c*g) / det;
    o[5] = -(a*f - c*d) / det;
    o[6] =  (d*h - e*g) / det;
    o[7] = -(a*h - b*g) / det;
    o[8] =  (a*e - b*d) / det;
}

__global__ __launch_bounds__(256) void lss_kernel(
        const float* __restrict__ x,
        const float* __restrict__ intrins,
        const float* __restrict__ post_rots,
        const float* __restrict__ post_trans,
        float* __restrict__ out) {
    __shared__ float s_feat[TP * FPITCH];     // [pixel][channel], padded
    __shared__ float s_depth[NPTS_TILE];      // exp values -> normalized depth
    __shared__ int   s_vox[NPTS_TILE];        // output base offset or -1
    __shared__ float s_den[TP];               // softmax denominators
    __shared__ float s_inv[18];               // invR(9) + invK(9)
    __shared__ float s_tr[3];                 // post_trans[b]

    const int t        = threadIdx.x;
    const int b        = blockIdx.x / TILES_PER_B;
    const int tile     = blockIdx.x % TILES_PER_B;
    const int pixbase  = tile * TP;
    const float* xb    = x + (size_t)b * TOTCH * NPIX_PER_B;

    if (t == 32) {  // wave 1 lane 0: tiny 3x3 inverses while wave 0 does softmax
        inv3(post_rots + b * 9, s_inv);
        inv3(intrins  + b * 9, s_inv + 9);
        s_tr[0] = post_trans[b*3+0];
        s_tr[1] = post_trans[b*3+1];
        s_tr[2] = post_trans[b*3+2];
    }

    if (t < 32) {
        // Wave 0: softmax over 41 depth logits for 16 pixels.
        // WMMA 16x16x4 f32 A layout: lanes 0-15 hold K=0,1; lanes 16-31 hold K=2,3.
        const int lane = t;
        const int j    = lane & 15;     // pixel within tile == M row
        const int h0   = lane >> 4;     // K half
        const float* px = xb + pixbase + j;

        float vals[22];
        float pmax = -INFINITY;
        #pragma unroll
        for (int it = 0; it < 11; ++it) {
            int k0 = it * 4 + h0 * 2;
            float a0 = (k0     < DPL) ? px[(size_t)(k0    ) * NPIX_PER_B] : -INFINITY;
            float a1 = (k0 + 1 < DPL) ? px[(size_t)(k0 + 1) * NPIX_PER_B] : -INFINITY;
            vals[it*2]   = a0;
            vals[it*2+1] = a1;
            pmax = fmaxf(pmax, fmaxf(a0, a1));
        }
        float m = fmaxf(pmax, __shfl_xor(pmax, 16, 32));  // combine K-halves per pixel

        v8f acc = {0.f,0.f,0.f,0.f,0.f,0.f,0.f,0.f};
        v2f bones = {1.0f, 1.0f};   // all-ones B => C columns = row sums of A
        #pragma unroll
        for (int it = 0; it < 11; ++it) {
            int k0 = it * 4 + h0 * 2;
            float e0 = (k0     < DPL) ? expf(vals[it*2]   - m) : 0.0f;
            float e1 = (k0 + 1 < DPL) ? expf(vals[it*2+1] - m) : 0.0f;
            if (k0     < DPL) s_depth[(k0    ) * TP + j] = e0;
            if (k0 + 1 < DPL) s_depth[(k0 + 1) * TP + j] = e1;
            v2f a = {e0, e1};
            acc = __builtin_amdgcn_wmma_f32_16x16x4_f32(
                      false, a, false, bones, (short)0, acc, false, false);
        }
        // C/D layout: VGPR v holds M=v (lanes 0-15) and M=v+8 (lanes 16-31).
        if (lane == 0) {
            s_den[0]=acc[0]; s_den[1]=acc[1]; s_den[2]=acc[2]; s_den[3]=acc[3];
            s_den[4]=acc[4]; s_den[5]=acc[5]; s_den[6]=acc[6]; s_den[7]=acc[7];
        } else if (lane == 16) {
            s_den[8]=acc[0];  s_den[9]=acc[1];  s_den[10]=acc[2]; s_den[11]=acc[3];
            s_den[12]=acc[4]; s_den[13]=acc[5]; s_den[14]=acc[6]; s_den[15]=acc[7];
        }
    }

    // All threads: stage the 16x64 feature tile (coalesced global reads).
    {
        const float* fb = xb + (size_t)DPL * NPIX_PER_B + pixbase;
        #pragma unroll
        for (int r = 0; r < (TP * NCH) / 256; ++r) {
            int idx = r * 256 + t;
            int c = idx >> 4, j = idx & 15;
            s_feat[j * FPITCH + c] = fb[(size_t)c * NPIX_PER_B + j];
        }
    }

    __syncthreads();

    // Geometry -> voxel index, plus depth normalization.
    for (int e = t; e < NPTS_TILE; e += 256) {
        int d = e >> 4, j = e & 15;
        int pix = pixbase + j;
        int h = pix / FW, w = pix - h * FW;
        float f0 = (float)w * (703.0f / 87.0f) - s_tr[0];
        float f1 = (float)h * (255.0f / 31.0f) - s_tr[1];
        float f2 = (4.0f + (float)d) - s_tr[2];
        float r0 = s_inv[0]*f0 + s_inv[1]*f1 + s_inv[2]*f2;
        float r1 = s_inv[3]*f0 + s_inv[4]*f1 + s_inv[5]*f2;
        float r2 = s_inv[6]*f0 + s_inv[7]*f1 + s_inv[8]*f2;
        float q0 = r0 * r2, q1 = r1 * r2;
        float g0 = s_inv[9]*q0  + s_inv[10]*q1 + s_inv[11]*r2;
        float g1 = s_inv[12]*q0 + s_inv[13]*q1 + s_inv[14]*r2;
        float g2 = s_inv[15]*q0 + s_inv[16]*q1 + s_inv[17]*r2;
        // trunc-toward-zero exactly matches jnp .astype(int32)
        int cx = (int)((g0 + 50.0f) / 0.5f);
        int cy = (int)((g1 + 10.0f) / 20.0f);
        int cz = (int)( g2          / 0.25f);
        bool kept = (cx >= 0) & (cx < NX0) & (cy >= 0) & (cy < 1) &
                    (cz >= 0) & (cz < NZ);
        s_vox[e] = kept ? (b * OUT_PER_B + cz * NX0 + cx) : -1;
        s_depth[e] = s_depth[e] / s_den[j];
    }

    __syncthreads();

    // Scatter: each wave = one point x 32 consecutive channels (broadcast LDS reads).
    #pragma unroll 4
    for (int it = 0; it < (NPTS_TILE * NCH) / 256; ++it) {
        int idx = it * 256 + t;
        int p = idx >> 6;       // point entry (d*16 + j)
        int c = idx & 63;       // channel
        int vox = s_vox[p];
        if (vox >= 0) {
            float val = s_depth[p] * s_feat[(p & 15) * FPITCH + c];
            atomicAdd(out + vox + c * CH_STRIDE, val);
        }
    }
}

extern "C" void kernel_launch(void* const* d_in, const int* in_sizes, int n_in,
                              void* d_out, int out_size, void* d_ws, size_t ws_size,
                              hipStream_t stream) {
    (void)in_sizes; (void)n_in; (void)d_ws; (void)ws_size;
    const float* x          = (const float*)d_in[0];
    const float* intrins    = (const float*)d_in[1];
    const float* post_rots  = (const float*)d_in[2];
    const float* post_trans = (const float*)d_in[3];
    float* out = (float*)d_out;

    int n4 = out_size / 4;  // 10,240,000 / 4
    zero_kernel<<<(n4 + 255) / 256, 256, 0, stream>>>((float4*)out, n4);
    lss_kernel<<<BATCH * TILES_PER_B, 256, 0, stream>>>(x, intrins, post_rots,
                                                        post_trans, out);
}